// Model_2637109919789
// MI455X (gfx1250) — compile-verified
//
#include <hip/hip_runtime.h>
#include <hip/hip_bf16.h>

#define N_CHEM 100000
#define N_PROT 20000
#define DIM    128
#define NEDGE  600000
#define NLBL   200000
#define NLAYER 3

typedef __attribute__((ext_vector_type(2))) float v2f;
typedef __attribute__((ext_vector_type(8))) float v8f;

// ---------------------------------------------------------------------------
// Zero-fill (float4 granularity, grid-stride)
// ---------------------------------------------------------------------------
__global__ void zero4_kernel(float4* __restrict__ p, long n4) {
    long i = (long)blockIdx.x * blockDim.x + threadIdx.x;
    long stride = (long)gridDim.x * blockDim.x;
    float4 z = {0.f, 0.f, 0.f, 0.f};
    for (; i < n4; i += stride) p[i] = z;
}

// ---------------------------------------------------------------------------
// Fused bidirectional edge scatter:
//   agg_p[dst] += h_c[src];  deg_p[dst] += 1
//   agg_c[src] += h_p[dst];  deg_c[src] += 1
// One wave (32 lanes) per edge, 4 floats per lane (D=128).
// ---------------------------------------------------------------------------
__global__ void scatter_kernel(const float* __restrict__ hc,
                               const float* __restrict__ hp,
                               const int*   __restrict__ src,
                               const int*   __restrict__ dst,
                               float* __restrict__ agg_c,
                               float* __restrict__ agg_p,
                               float* __restrict__ deg_c,
                               float* __restrict__ deg_p) {
    int lane = threadIdx.x & 31;
    int e = (int)(((long)blockIdx.x * blockDim.x + threadIdx.x) >> 5);
    if (e >= NEDGE) return;
    int s = src[e];
    int d = dst[e];
    const float4 vc = *(const float4*)(hc + (long)s * DIM + lane * 4);
    const float4 vp = *(const float4*)(hp + (long)d * DIM + lane * 4);
    float* ap = agg_p + (long)d * DIM + lane * 4;
    float* ac = agg_c + (long)s * DIM + lane * 4;
    atomicAdd(ap + 0, vc.x); atomicAdd(ap + 1, vc.y);
    atomicAdd(ap + 2, vc.z); atomicAdd(ap + 3, vc.w);
    atomicAdd(ac + 0, vp.x); atomicAdd(ac + 1, vp.y);
    atomicAdd(ac + 2, vp.z); atomicAdd(ac + 3, vp.w);
    if (lane == 0) {
        atomicAdd(deg_p + d, 1.f);
        atomicAdd(deg_c + s, 1.f);
    }
}

// ---------------------------------------------------------------------------
// Fused SAGE layer GEMM (full fp32 via V_WMMA_F32_16X16X4_F32):
//   out = relu( (agg/clip(deg,1)) @ Wl.T + bl + hdst @ Wr.T )
// One wave per 16x16 output tile; K stepped by 4 over 128.
// A 16x4 frag: lane&15 = M row, (lane>>4)*2 = K offset, VGPR0/1 = K,K+1.
// B 4x16 frag: lane&15 = N col (weight row j), same K offset scheme.
// C/D: VGPR v holds M = v + 8*(lane>>4), N = lane&15.
// ---------------------------------------------------------------------------
__global__ void sage_gemm_kernel(const float* __restrict__ agg,   // [n, D]
                                 const float* __restrict__ deg,   // [n]
                                 const float* __restrict__ hdst,  // [n, D]
                                 const float* __restrict__ Wl,    // [D, D] (j,k)
                                 const float* __restrict__ bl,    // [D]
                                 const float* __restrict__ Wr,    // [D, D] (j,k)
                                 float* __restrict__ out)         // [n, D]
{
    int wave = (int)(((long)blockIdx.x * blockDim.x + threadIdx.x) >> 5);
    int lane = threadIdx.x & 31;
    int tileN = wave & 7;      // DIM/16 = 8 column tiles
    int tileM = wave >> 3;
    int m0 = tileM * 16;
    int n0 = tileN * 16;
    int lo = lane & 15;
    int hi = lane >> 4;        // 0 or 1
    int koff = hi * 2;

    int rowA = m0 + lo;        // fixed per lane for A fragments
    int colB = n0 + lo;        // fixed per lane for B fragments
    float rdeg = 1.0f / fmaxf(deg[rowA], 1.0f);

    const float* arow  = agg  + (long)rowA * DIM;
    const float* hrow  = hdst + (long)rowA * DIM;
    const float* wlcol = Wl   + (long)colB * DIM;
    const float* wrcol = Wr   + (long)colB * DIM;

    v8f acc = {};
    // mean @ Wl.T
#pragma unroll
    for (int k0 = 0; k0 < DIM; k0 += 4) {
        int kb = k0 + koff;
        v2f a, b;
        a.x = arow[kb]     * rdeg;
        a.y = arow[kb + 1] * rdeg;
        b.x = wlcol[kb];
        b.y = wlcol[kb + 1];
        acc = __builtin_amdgcn_wmma_f32_16x16x4_f32(false, a, false, b,
                                                    (short)0, acc, false, false);
    }
    // + hdst @ Wr.T
#pragma unroll
    for (int k0 = 0; k0 < DIM; k0 += 4) {
        int kb = k0 + koff;
        v2f a, b;
        a.x = hrow[kb];
        a.y = hrow[kb + 1];
        b.x = wrcol[kb];
        b.y = wrcol[kb + 1];
        acc = __builtin_amdgcn_wmma_f32_16x16x4_f32(false, a, false, b,
                                                    (short)0, acc, false, false);
    }

    float bias = bl[colB];
#pragma unroll
    for (int v = 0; v < 8; ++v) {
        int r = m0 + v + hi * 8;
        float val = acc[v] + bias;
        out[(long)r * DIM + n0 + lo] = fmaxf(val, 0.0f);
    }
}

// ---------------------------------------------------------------------------
// Edge decoder: x = concat(h_c[row], h_p[col]); z = x @ Wd.T + bd
// One 256-thread block per label edge.
// ---------------------------------------------------------------------------
__global__ void decoder_kernel(const float* __restrict__ hc,
                               const float* __restrict__ hp,
                               const int*   __restrict__ row,
                               const int*   __restrict__ col,
                               const float* __restrict__ Wd,   // [2D]
                               const float* __restrict__ bd,   // [1]
                               float* __restrict__ z,          // [EL]
                               float* __restrict__ x)          // [EL, 2D]
{
    int e = blockIdx.x;
    int t = threadIdx.x;    // 0..255
    int r = row[e];
    int c = col[e];
    float v = (t < DIM) ? hc[(long)r * DIM + t]
                        : hp[(long)c * DIM + (t - DIM)];
    x[(long)e * (2 * DIM) + t] = v;

    float p = v * Wd[t];
    // wave32 reduce
#pragma unroll
    for (int off = 16; off > 0; off >>= 1)
        p += __shfl_down(p, off, 32);
    __shared__ float red[8];
    if ((t & 31) == 0) red[t >> 5] = p;
    __syncthreads();
    if (t == 0) {
        float s = 0.f;
#pragma unroll
        for (int i = 0; i < 8; ++i) s += red[i];
        z[e] = s + bd[0];
    }
}

// ---------------------------------------------------------------------------
extern "C" void kernel_launch(void* const* d_in, const int* in_sizes, int n_in,
                              void* d_out, int out_size, void* d_ws, size_t ws_size,
                              hipStream_t stream) {
    (void)in_sizes; (void)n_in; (void)out_size; (void)ws_size;

    const float* x_chem = (const float*)d_in[0];
    const float* x_prot = (const float*)d_in[1];
    const float* Wl     = (const float*)d_in[2];   // [L,2,D,D]
    const float* bl     = (const float*)d_in[3];   // [L,2,D]
    const float* Wr     = (const float*)d_in[4];   // [L,2,D,D]
    const float* Wd     = (const float*)d_in[5];   // [1,2D]
    const float* bd     = (const float*)d_in[6];   // [1]
    const int*   src    = (const int*)d_in[7];
    const int*   dst    = (const int*)d_in[8];
    const int*   rowi   = (const int*)d_in[9];
    const int*   coli   = (const int*)d_in[10];

    // workspace layout (floats): hc0, hc1, hp0, hp1, agg_c, agg_p, deg_c, deg_p
    float* ws = (float*)d_ws;
    float* hc0   = ws;  ws += (long)N_CHEM * DIM;
    float* hc1   = ws;  ws += (long)N_CHEM * DIM;
    float* hp0   = ws;  ws += (long)N_PROT * DIM;
    float* hp1   = ws;  ws += (long)N_PROT * DIM;
    float* agg_c = ws;  ws += (long)N_CHEM * DIM;
    float* agg_p = ws;  ws += (long)N_PROT * DIM;
    float* deg_c = ws;  ws += N_CHEM;
    float* deg_p = ws;  ws += N_PROT;

    hipMemcpyAsync(hc0, x_chem, sizeof(float) * (size_t)N_CHEM * DIM,
                   hipMemcpyDeviceToDevice, stream);
    hipMemcpyAsync(hp0, x_prot, sizeof(float) * (size_t)N_PROT * DIM,
                   hipMemcpyDeviceToDevice, stream);

    float* hc_in = hc0; float* hc_out = hc1;
    float* hp_in = hp0; float* hp_out = hp1;

    const long zero_n4 =
        ((long)N_CHEM * DIM + (long)N_PROT * DIM + N_CHEM + N_PROT) / 4;
    const int scatter_blocks = (NEDGE * 32) / 256;           // exact: 75000
    const int prot_blocks = (N_PROT / 16) * 8 / 4;           // 10000 waves / 4
    const int chem_blocks = (N_CHEM / 16) * 8 / 4;           // 50000 waves / 4

    for (int l = 0; l < NLAYER; ++l) {
        // agg_c/agg_p/deg_c/deg_p are contiguous -> single zero pass
        zero4_kernel<<<2048, 256, 0, stream>>>((float4*)agg_c, zero_n4);

        scatter_kernel<<<scatter_blocks, 256, 0, stream>>>(
            hc_in, hp_in, src, dst, agg_c, agg_p, deg_c, deg_p);

        // relation 0: chem->protein (dst protein)
        {
            const float* Wl0 = Wl + ((long)l * 2 + 0) * DIM * DIM;
            const float* bl0 = bl + ((long)l * 2 + 0) * DIM;
            const float* Wr0 = Wr + ((long)l * 2 + 0) * DIM * DIM;
            sage_gemm_kernel<<<prot_blocks, 128, 0, stream>>>(
                agg_p, deg_p, hp_in, Wl0, bl0, Wr0, hp_out);
        }
        // relation 1: protein->chem (dst chem)
        {
            const float* Wl1 = Wl + ((long)l * 2 + 1) * DIM * DIM;
            const float* bl1 = bl + ((long)l * 2 + 1) * DIM;
            const float* Wr1 = Wr + ((long)l * 2 + 1) * DIM * DIM;
            sage_gemm_kernel<<<chem_blocks, 128, 0, stream>>>(
                agg_c, deg_c, hc_in, Wl1, bl1, Wr1, hc_out);
        }
        // ping-pong
        float* t;
        t = hc_in; hc_in = hc_out; hc_out = t;
        t = hp_in; hp_in = hp_out; hp_out = t;
    }

    float* z    = (float*)d_out;          // [EL]
    float* xcat = (float*)d_out + NLBL;   // [EL, 2D]
    decoder_kernel<<<NLBL, 256, 0, stream>>>(hc_in, hp_in, rowi, coli,
                                             Wd, bd, z, xcat);
}